// GroupedEmbedding_72241349918733
// MI455X (gfx1250) — compile-verified
//
#include <hip/hip_runtime.h>

// ---------------------------------------------------------------------------
// GroupedEmbedding == single gather: out[t, :] = weight[idx[t], :]
//   idx:    int32[32768]  (B=8, S=4096 flattened)
//   weight: float32[128000, 1024]
//   out:    float32[32768, 1024]
// Pure bandwidth problem (~256 MB @ 23.3 TB/s ~ 11 us). One wave32 per two 4 KB
// rows; rows staged through LDS with CDNA5 async copies (ASYNCcnt), bypassing
// VGPRs. Structure: issue ALL row loads for the wave's tokens -> single
// s_wait_asynccnt 0 -> issue ALL stores, so each wave pays ~one memory
// latency total instead of one per token. Stores use TH=NT (write-once
// output, keep L2 free for weight rows).
// ---------------------------------------------------------------------------

#ifndef __has_builtin
#define __has_builtin(x) 0
#endif

#if defined(__HIP_DEVICE_COMPILE__) &&                                     \
    __has_builtin(__builtin_amdgcn_global_load_async_to_lds_b128) &&       \
    __has_builtin(__builtin_amdgcn_global_store_async_from_lds_b128)
#define EMB_USE_ASYNC_LDS 1
#else
#define EMB_USE_ASYNC_LDS 0
#endif

#if defined(__HIP_DEVICE_COMPILE__) && __has_builtin(__builtin_amdgcn_s_wait_asynccnt)
#define EMB_WAIT_ASYNC0() __builtin_amdgcn_s_wait_asynccnt(0)
#elif defined(__HIP_DEVICE_COMPILE__)
#define EMB_WAIT_ASYNC0() asm volatile("s_wait_asynccnt 0" ::: "memory")
#else
#define EMB_WAIT_ASYNC0()
#endif

// 16-byte vector type matching the builtin's parameter:
//   'int __attribute__((vector_size(16))) __device__ *'  (AS1 / AS3)
typedef int v4i __attribute__((vector_size(16)));
#if defined(__HIP_DEVICE_COMPILE__)
typedef __attribute__((address_space(1))) v4i* emb_gptr_t;  // global
typedef __attribute__((address_space(3))) v4i* emb_lptr_t;  // LDS
#endif

namespace {
constexpr int kDim       = 1024;        // floats per row
constexpr int kDim4      = kDim / 4;    // float4 per row = 256
constexpr int kChunks    = kDim4 / 32;  // 8 chunks: lane + c*32 covers a row
constexpr int kWaves     = 8;           // 256-thread block
constexpr int kTokPerWv  = 2;           // tokens staged per wave (8 KB LDS)
constexpr int kTokPerBlk = kWaves * kTokPerWv;  // 16
}  // namespace

__global__ __launch_bounds__(256) void GroupedEmbedding_gather_kernel(
    const int* __restrict__ idx, const float4* __restrict__ w4,
    float4* __restrict__ out4, int n_tokens) {
  const int lane = threadIdx.x & 31;
  const int wave = threadIdx.x >> 5;
  const int t0   = (blockIdx.x * kWaves + wave) * kTokPerWv;

#if EMB_USE_ASYNC_LDS
  __shared__ float4 lds4[kWaves * kTokPerWv * kDim4];  // 64 KB / block

  // Phase 1: issue all async row loads (memory -> LDS), disjoint buffers.
#pragma unroll
  for (int i = 0; i < kTokPerWv; ++i) {
    const int t = t0 + i;
    if (t >= n_tokens) break;  // wave-uniform
    const int row = __builtin_amdgcn_readfirstlane(idx[t]);
    const float4* __restrict__ gsrc = w4 + (size_t)row * kDim4 + lane;
    float4* lb = &lds4[(wave * kTokPerWv + i) * kDim4] + lane;
#pragma unroll
    for (int c = 0; c < kChunks; ++c) {
      __builtin_amdgcn_global_load_async_to_lds_b128(
          (emb_gptr_t)(gsrc + c * 32), (emb_lptr_t)(lb + c * 32),
          /*offset=*/0, /*cpol=*/0);
    }
  }
  EMB_WAIT_ASYNC0();  // single latency-hide point: all LDS writes complete

  // Phase 2: issue all async row stores (LDS -> memory).
#pragma unroll
  for (int i = 0; i < kTokPerWv; ++i) {
    const int t = t0 + i;
    if (t >= n_tokens) break;
    float4* __restrict__ gdst = out4 + (size_t)t * kDim4 + lane;
    const float4* lb = &lds4[(wave * kTokPerWv + i) * kDim4] + lane;
#pragma unroll
    for (int c = 0; c < kChunks; ++c) {
      __builtin_amdgcn_global_store_async_from_lds_b128(
          (emb_gptr_t)(gdst + c * 32), (emb_lptr_t)(lb + c * 32),
          /*offset=*/0, /*cpol=*/1);  // TH=NT: streaming, write-once output
    }
  }
  EMB_WAIT_ASYNC0();  // drained before S_ENDPGM (which also waits idle)
#else
  // Fallback: coalesced float4 register copy (global_load/store_b128).
  for (int i = 0; i < kTokPerWv; ++i) {
    const int t = t0 + i;
    if (t >= n_tokens) return;
    const int row = __builtin_amdgcn_readfirstlane(idx[t]);
    const float4* __restrict__ gsrc = w4 + (size_t)row * kDim4 + lane;
    float4* __restrict__ gdst       = out4 + (size_t)t * kDim4 + lane;
#pragma unroll
    for (int c = 0; c < kChunks; ++c) {
      gdst[c * 32] = gsrc[c * 32];
    }
  }
#endif
}

extern "C" void kernel_launch(void* const* d_in, const int* in_sizes, int n_in,
                              void* d_out, int out_size, void* d_ws,
                              size_t ws_size, hipStream_t stream) {
  (void)n_in; (void)out_size; (void)d_ws; (void)ws_size;
  const int* idx     = (const int*)d_in[0];     // input_  int32[B*S]
  const float4* w4   = (const float4*)d_in[1];  // weight  f32[128000,1024]
  float4* out4       = (float4*)d_out;          // out     f32[B*S,1024]
  const int n_tokens = in_sizes[0];             // 32768

  const int blocks = (n_tokens + kTokPerBlk - 1) / kTokPerBlk;  // 2048
  GroupedEmbedding_gather_kernel<<<blocks, 256, 0, stream>>>(idx, w4, out4,
                                                             n_tokens);
}